// GeometryEngine_85074712199907
// MI455X (gfx1250) — compile-verified
//
#include <hip/hip_runtime.h>

// Optimized Ricci-scalar-at-a-point kernel for MI455X (gfx1250).
//
// Key insight: the reference's output depends only on metric values at
// lattice points [active-2, active+2] (central finite differences, interior
// point). So instead of a 16MB / O(L*4^4) dense pipeline, we run a single
// wave32 workgroup that loads 320 bytes, inverts three 4x4 metrics, forms
// Christoffels, and contracts. The Gamma*Gamma Riemann term is computed as a
// genuine 16x4 @ 4x16 f32 matrix product with V_WMMA_F32_16X16X4_F32.

#define DXC 1.0f

typedef __attribute__((ext_vector_type(2))) float v2f;
typedef __attribute__((ext_vector_type(8))) float v8f;

// Standard cofactor 4x4 inverse (row-major in, row-major out).
__device__ __forceinline__ void inv4(const float* m, float* r) {
    float a00=m[0],  a01=m[1],  a02=m[2],  a03=m[3];
    float a10=m[4],  a11=m[5],  a12=m[6],  a13=m[7];
    float a20=m[8],  a21=m[9],  a22=m[10], a23=m[11];
    float a30=m[12], a31=m[13], a32=m[14], a33=m[15];
    float b00=a00*a11-a01*a10, b01=a00*a12-a02*a10, b02=a00*a13-a03*a10;
    float b03=a01*a12-a02*a11, b04=a01*a13-a03*a11, b05=a02*a13-a03*a12;
    float b06=a20*a31-a21*a30, b07=a20*a32-a22*a30, b08=a20*a33-a23*a30;
    float b09=a21*a32-a22*a31, b10=a21*a33-a23*a31, b11=a22*a33-a23*a32;
    float det = b00*b11 - b01*b10 + b02*b09 + b03*b08 - b04*b07 + b05*b06;
    float id = 1.0f / det;
    r[0]  = ( a11*b11 - a12*b10 + a13*b09)*id;
    r[1]  = (-a01*b11 + a02*b10 - a03*b09)*id;
    r[2]  = ( a31*b05 - a32*b04 + a33*b03)*id;
    r[3]  = (-a21*b05 + a22*b04 - a23*b03)*id;
    r[4]  = (-a10*b11 + a12*b08 - a13*b07)*id;
    r[5]  = ( a00*b11 - a02*b08 + a03*b07)*id;
    r[6]  = (-a30*b05 + a32*b02 - a33*b01)*id;
    r[7]  = ( a20*b05 - a22*b02 + a23*b01)*id;
    r[8]  = ( a10*b10 - a11*b08 + a13*b06)*id;
    r[9]  = (-a00*b10 + a01*b08 - a03*b06)*id;
    r[10] = ( a30*b04 - a31*b02 + a33*b00)*id;
    r[11] = (-a20*b04 + a21*b02 - a23*b00)*id;
    r[12] = (-a10*b09 + a11*b07 - a12*b06)*id;
    r[13] = ( a00*b09 - a01*b07 + a02*b06)*id;
    r[14] = (-a30*b03 + a31*b01 - a32*b00)*id;
    r[15] = ( a20*b03 - a21*b01 + a22*b00)*id;
}

__global__ __launch_bounds__(32)
void GeometryEngine_ricci_scalar_kernel(const float* __restrict__ gfield,
                                        const int* __restrict__ active,
                                        float* __restrict__ out) {
    // Local indexing: global lattice points (ai-2 .. ai+2) -> g[0..4].
    // g_inv / M / C arrays cover points (ai-1, ai, ai+1) -> index 0..2.
    __shared__ float g[5][16];    // metric tiles
    __shared__ float gi[3][16];   // inverse metric
    __shared__ float Mm[3][16];   // M = g_inv @ dg   (dg = d/dx of metric)
    __shared__ float C[3][64];    // Christoffel: C[p][a*16 + m*4 + n]
    __shared__ float GG[16][16];  // WMMA result: GG[a*4+m][b*4+n]

    const int tid = threadIdx.x;
    const long long ai = (long long)active[0];  // interior: central diffs valid

    // ---- Load the 5-point metric neighborhood (320 bytes total) ----
    for (int e = tid; e < 80; e += 32) {
        int p = e >> 4, q = e & 15;
        g[p][q] = gfield[(ai - 2 + p) * 16 + q];
    }
    __syncthreads();

    // ---- 4x4 inverses at the three points that feed Christoffels ----
    if (tid < 3) inv4(&g[tid + 1][0], &gi[tid][0]);
    __syncthreads();

    const float inv2dx = 0.5f / DXC;

    // ---- M_p = g_inv_p @ sg_p, sg_p = central diff of metric (48 entries) ----
    for (int e = tid; e < 48; e += 32) {
        int p = e >> 4, an = e & 15, a = an >> 2, n = an & 3;
        float acc = 0.0f;
        #pragma unroll
        for (int s = 0; s < 4; ++s) {
            float sg = (g[p + 2][s * 4 + n] - g[p][s * 4 + n]) * inv2dx;
            acc += gi[p][a * 4 + s] * sg;
        }
        Mm[p][an] = acc;
    }
    __syncthreads();

    // ---- Christoffels at the three points (192 entries) ----
    // C[a,m,n] = 0.5*( [m==1]*M[a,n] + [n==1]*M[a,m] - g_inv[a,1]*sg[m,n] )
    for (int e = tid; e < 192; e += 32) {
        int p = e >> 6, r = e & 63;
        int a = r >> 4, m = (r >> 2) & 3, n = r & 3;
        float sg_mn = (g[p + 2][m * 4 + n] - g[p][m * 4 + n]) * inv2dx;
        float v = -gi[p][a * 4 + 1] * sg_mn;
        if (m == 1) v += Mm[p][a * 4 + n];
        if (n == 1) v += Mm[p][a * 4 + m];
        C[p][r] = 0.5f * v;
    }
    __syncthreads();

    // ---- Gamma*Gamma term via V_WMMA_F32_16X16X4_F32 ----
    // GG[(a,m),(b,n)] = sum_s P[(a,m),s] * Q[s,(b,n)]
    //   P[(a,m),s] = C[a,s,m]   (A matrix, 16x4)
    //   Q[s,(b,n)] = C[s,b,n]   (B matrix, 4x16)
    // A layout (ISA 7.12.2, f32 16x4): lane = M (mod 16); VGPR0 = K0/K2,
    // VGPR1 = K1/K3 for low/high lane halves. B symmetric. All 32 lanes active.
    {
        const int row = tid & 15;       // M for A-pack, N for B-pack, col for D
        const int hi  = tid >> 4;       // 0: K={0,1}; 1: K={2,3}
        const int a = row >> 2, m = row & 3;
        v2f A, B;
        A.x = C[1][a * 16 + (2 * hi + 0) * 4 + m];
        A.y = C[1][a * 16 + (2 * hi + 1) * 4 + m];
        B.x = C[1][(2 * hi + 0) * 16 + row];
        B.y = C[1][(2 * hi + 1) * 16 + row];
        v8f Dacc = {};
        Dacc = __builtin_amdgcn_wmma_f32_16x16x4_f32(
            /*neg_a=*/false, A, /*neg_b=*/false, B,
            /*c_mod=*/(short)0, Dacc, /*reuse_a=*/false, /*reuse_b=*/false);
        // D layout: VGPR v -> M = v (lanes 0-15) / 8+v (lanes 16-31), N = lane&15
        const int rbase = hi * 8;
        #pragma unroll
        for (int v = 0; v < 8; ++v) GG[rbase + v][row] = Dacc[v];
    }
    __syncthreads();

    // ---- Final contractions (tiny; single lane) ----
    // scalar = sum_{b,n} gi[b,n] * ( dC[1,b,n]
    //                              + sum_a (GG[a,b,a,n] - GG[a,b,n,a]) )
    //        - sum_b gi[b,1] * sum_a dC[a,b,a]
    if (tid == 0) {
        float s = 0.0f;
        #pragma unroll
        for (int b = 0; b < 4; ++b) {
            #pragma unroll
            for (int n = 0; n < 4; ++n) {
                float dc1 = (C[2][16 + b * 4 + n] - C[0][16 + b * 4 + n]) * inv2dx;
                float acc = dc1;
                #pragma unroll
                for (int a = 0; a < 4; ++a)
                    acc += GG[a * 5][b * 4 + n] - GG[a * 4 + n][b * 4 + a];
                s += gi[1][b * 4 + n] * acc;
            }
            float tr = 0.0f;
            #pragma unroll
            for (int a = 0; a < 4; ++a)
                tr += (C[2][a * 16 + b * 4 + a] - C[0][a * 16 + b * 4 + a]) * inv2dx;
            s -= gi[1][b * 4 + 1] * tr;
        }
        out[0] = s;
    }
}

extern "C" void kernel_launch(void* const* d_in, const int* in_sizes, int n_in,
                              void* d_out, int out_size, void* d_ws, size_t ws_size,
                              hipStream_t stream) {
    const float* gfield = (const float*)d_in[0];   // [L,4,4] float32
    const int*   active = (const int*)d_in[1];     // scalar int
    float*       out    = (float*)d_out;           // 1 float
    (void)in_sizes; (void)n_in; (void)out_size; (void)d_ws; (void)ws_size;
    GeometryEngine_ricci_scalar_kernel<<<1, 32, 0, stream>>>(gfield, active, out);
}